// VertexSplitter_63015760167455
// MI455X (gfx1250) — compile-verified
//
#include <hip/hip_runtime.h>
#include <cstdint>
#include <cstddef>

// Problem constants from the reference: B=256, V=512.
#define BATCH 256
#define VDIM  512

typedef __attribute__((ext_vector_type(4))) float v4f;

// ---------------------------------------------------------------------------
// Phase 1: streaming threshold  out[i] = Pid[i] > 0 ? 1 : 0, duplicated into
// both output halves. Memory-bound: 256MB read + 512MB write ~ 33us @ 23.3TB/s.
// Uses the CDNA5 async global->LDS path (ASYNCcnt) for the load side and
// non-temporal b128 stores so the write-once output doesn't pollute L2.
// ---------------------------------------------------------------------------
__global__ __launch_bounds__(256) void bulk_threshold_kernel(
    const float* __restrict__ in, float* __restrict__ out,
    size_t n4 /* total float4 elements per half */) {
  __shared__ v4f tile[256];

  size_t idx = (size_t)blockIdx.x * 256u + threadIdx.x;
  if (idx >= n4) return;

  const v4f* gin = (const v4f*)in;
  uint64_t gaddr = (uint64_t)(size_t)(gin + idx);
  // Generic pointer to LDS: low 32 bits are the wave-relative LDS byte offset.
  uint32_t lds_off = (uint32_t)(size_t)&tile[threadIdx.x];

  // CDNA5 async copy: global memory -> LDS, tracked by ASYNCcnt.
  asm volatile("global_load_async_to_lds_b128 %0, %1, off"
               :: "v"(lds_off), "v"(gaddr)
               : "memory");
  asm volatile("s_wait_asynccnt 0" ::: "memory");

  v4f v = tile[threadIdx.x];  // ds_load_b128 of this lane's own slot
  v4f r;
  r.x = (v.x > 0.0f) ? 1.0f : 0.0f;
  r.y = (v.y > 0.0f) ? 1.0f : 0.0f;
  r.z = (v.z > 0.0f) ? 1.0f : 0.0f;
  r.w = (v.w > 0.0f) ? 1.0f : 0.0f;

  v4f* o = (v4f*)out;
  __builtin_nontemporal_store(r, o + idx);        // half 0
  __builtin_nontemporal_store(r, o + n4 + idx);   // half 1 (duplicate output)
}

// ---------------------------------------------------------------------------
// Phase 2: per-batch scatter edits. One thread per batch (256 total).
// Reads M (= thresholded half 0, pre-edit) for blocked/old_pid, then applies
// 8 distinct single-element writes to both halves when valid.
// ---------------------------------------------------------------------------
__global__ __launch_bounds__(256) void fixup_kernel(
    const int* __restrict__ inter, float* __restrict__ out,
    size_t plane /* V*V */, size_t half /* B*V*V */) {
  int b = (int)(blockIdx.x * 256u + threadIdx.x);
  if (b >= BATCH) return;

  int e10 = inter[4 * b + 0];
  int e11 = inter[4 * b + 1];
  int e20 = inter[4 * b + 2];
  int e21 = inter[4 * b + 3];

  bool distinct4 = (e10 != e11) && (e10 != e20) && (e10 != e21) &&
                   (e11 != e20) && (e11 != e21) && (e20 != e21);

  float* M0 = out + (size_t)b * plane;   // half 0 plane for this batch
  float* M1 = M0 + half;                 // half 1 plane

  float m_a = M0[(size_t)e10 * VDIM + e20];
  float m_b = M0[(size_t)e11 * VDIM + e21];
  float old_pid = M0[(size_t)e10 * VDIM + e11];

  bool blocked = (m_a == 1.0f) || (m_b == 1.0f);
  bool valid = distinct4 && !blocked;

  if (valid) {
    // All 8 ordered pairs are distinct because e10,e11,e20,e21 are distinct,
    // so write order is irrelevant.
    size_t i;
    i = (size_t)e10 * VDIM + e11; M0[i] = 0.0f;    M1[i] = 0.0f;
    i = (size_t)e11 * VDIM + e10; M0[i] = 0.0f;    M1[i] = 0.0f;
    i = (size_t)e20 * VDIM + e21; M0[i] = 0.0f;    M1[i] = 0.0f;
    i = (size_t)e21 * VDIM + e20; M0[i] = 0.0f;    M1[i] = 0.0f;
    i = (size_t)e10 * VDIM + e20; M0[i] = old_pid; M1[i] = old_pid;
    i = (size_t)e20 * VDIM + e10; M0[i] = old_pid; M1[i] = old_pid;
    i = (size_t)e11 * VDIM + e21; M0[i] = 1.0f;    M1[i] = 1.0f;
    i = (size_t)e21 * VDIM + e11; M0[i] = 1.0f;    M1[i] = 1.0f;
  }
}

extern "C" void kernel_launch(void* const* d_in, const int* in_sizes, int n_in,
                              void* d_out, int out_size, void* d_ws, size_t ws_size,
                              hipStream_t stream) {
  (void)in_sizes; (void)n_in; (void)out_size; (void)d_ws; (void)ws_size;

  const float* Pid = (const float*)d_in[0];          // (256, 512, 512) fp32
  const int* intersections = (const int*)d_in[1];    // (256, 2, 2) int32
  float* out = (float*)d_out;                        // 2 x (256*512*512) fp32

  const size_t total = (size_t)BATCH * VDIM * VDIM;  // 67,108,864 floats/half
  const size_t n4 = total / 4;                       // 16,777,216 float4/half

  dim3 grid((unsigned)((n4 + 255) / 256));
  bulk_threshold_kernel<<<grid, 256, 0, stream>>>(Pid, out, n4);
  fixup_kernel<<<1, 256, 0, stream>>>(intersections, out,
                                      (size_t)VDIM * VDIM, total);
}